// FeaturelessHeteroGAT_87677462381163
// MI455X (gfx1250) — compile-verified
//
#include <hip/hip_runtime.h>
#include <hip/hip_bf16.h>
#include <math.h>

// CDNA5 WMMA fp32 fragments
typedef float v2f __attribute__((ext_vector_type(2)));
typedef float v8f __attribute__((ext_vector_type(8)));

#define HDS 64   // hidden size
#define NH  4    // heads
#define HDD 16   // head dim

// ---------------- utility kernels ----------------

__global__ void k_fill(float* __restrict__ p, float v, long long n) {
    long long t = (long long)blockIdx.x * blockDim.x + threadIdx.x;
    if (t < n) p[t] = v;
}

__global__ void k_degree(const int* __restrict__ dst, float* __restrict__ deg, int E_) {
    int e = blockIdx.x * blockDim.x + threadIdx.x;
    if (e < E_) atomicAdd(&deg[dst[e]], 1.0f);
}

// 64 threads: the tiny emb @ W matvecs that layer 1 collapses to.
__global__ void k_small_vecs(const float* __restrict__ embW, const float* __restrict__ embT,
                             const float* __restrict__ WlWT, const float* __restrict__ WrWT, const float* __restrict__ blWT,
                             const float* __restrict__ WlTW, const float* __restrict__ WrTW, const float* __restrict__ blTW,
                             const float* __restrict__ WlWW, const float* __restrict__ WrWW, const float* __restrict__ blWW,
                             float* __restrict__ vecs) {
    int j = threadIdx.x;
    if (j >= 64) return;
    float vta = 0.f, vtc = 0.f, va1 = 0.f, va2 = 0.f, vwc = 0.f;
    for (int i = 0; i < 64; ++i) {
        float ew = embW[i], et = embT[i];
        vta += ew * WlWT[i * 64 + j];
        vtc += et * WrWT[i * 64 + j];
        va1 += et * WlTW[i * 64 + j];
        va2 += ew * WlWW[i * 64 + j];
        vwc += ew * (WrTW[i * 64 + j] + WrWW[i * 64 + j]);
    }
    vecs[j]        = vta;                       // vt_a
    vecs[64 + j]   = vtc + blWT[j];             // vt_c
    vecs[128 + j]  = va1;                       // vw_a1
    vecs[192 + j]  = va2;                       // vw_a2
    vecs[256 + j]  = vwc + blTW[j] + blWW[j];   // vw_c
}

// Layer 1: h = elu(layernorm(deg1*a1 (+ deg2*a2) + c)). One wave32 per node,
// each lane owns elements (lane, lane+32); reductions via shfl_xor.
__global__ void k_layer1(const float* __restrict__ deg1, const float* __restrict__ deg2,
                         const float* __restrict__ a1, const float* __restrict__ a2,
                         const float* __restrict__ cvec,
                         const float* __restrict__ g, const float* __restrict__ b,
                         float* __restrict__ hout, int n) {
    int node = blockIdx.x * (blockDim.x >> 5) + (threadIdx.x >> 5);
    int lane = threadIdx.x & 31;
    if (node >= n) return;
    float d1 = deg1[node];
    int j0 = lane, j1 = lane + 32;
    float x0 = d1 * a1[j0] + cvec[j0];
    float x1 = d1 * a1[j1] + cvec[j1];
    if (a2) {
        float d2 = deg2[node];
        x0 += d2 * a2[j0];
        x1 += d2 * a2[j1];
    }
    float s = x0 + x1;
    for (int m = 16; m; m >>= 1) s += __shfl_xor(s, m, 32);
    float mean = s * (1.0f / 64.0f);
    float e0 = x0 - mean, e1 = x1 - mean;
    float v = e0 * e0 + e1 * e1;
    for (int m = 16; m; m >>= 1) v += __shfl_xor(v, m, 32);
    float inv = rsqrtf(v * (1.0f / 64.0f) + 1e-5f);
    float y0 = e0 * inv * g[j0] + b[j0];
    float y1 = e1 * inv * g[j1] + b[j1];
    hout[(size_t)node * 64 + j0] = (y0 > 0.f) ? y0 : expm1f(y0);
    hout[(size_t)node * 64 + j1] = (y1 > 0.f) ? y1 : expm1f(y1);
}

// ---------------- WMMA projection kernel ----------------
// X = Hin(Nx64) @ W(64x64) via v_wmma_f32_16x16x4_f32; fused per-head attention
// score Aout[n,h] = sum_d X[n,h*16+d]*avec[h*16+d] (head == 16-col tile).
// Block = 128 threads = 4 waves; one 16-row tile per wave. W staged in LDS
// with row stride 65 to avoid bank conflicts between the two half-waves.
__global__ void k_project(const float* __restrict__ Hin, const float* __restrict__ W,
                          const float* __restrict__ avec,
                          float* __restrict__ Xout, float* __restrict__ Aout, int n) {
    __shared__ float Wl[64 * 65];
    __shared__ float Al[64];
    for (int i = threadIdx.x; i < 4096; i += blockDim.x) {
        int r = i >> 6, c = i & 63;
        Wl[r * 65 + c] = W[i];
    }
    if (threadIdx.x < 64) Al[threadIdx.x] = avec[threadIdx.x];
    __syncthreads();

    int lane = threadIdx.x & 31;
    int wv   = threadIdx.x >> 5;
    int tile = blockIdx.x * 4 + wv;              // 16-row tile index
    int row  = tile * 16 + (lane & 15);
    int kp   = (lane >> 4) * 2;                  // K sub-pair: 0 (lanes 0-15) / 2 (lanes 16-31)
    bool rowOK = row < n;

    // A fragments for all 16 K-steps (K = 64 = 16 * 4)
    v2f afrag[16];
#pragma unroll
    for (int kk = 0; kk < 16; ++kk) {
        afrag[kk].x = rowOK ? Hin[(size_t)row * 64 + kk * 4 + kp]     : 0.0f;
        afrag[kk].y = rowOK ? Hin[(size_t)row * 64 + kk * 4 + kp + 1] : 0.0f;
    }

#pragma unroll
    for (int nt = 0; nt < 4; ++nt) {             // N tile == head index
        int col = nt * 16 + (lane & 15);
        v8f acc = {0.f, 0.f, 0.f, 0.f, 0.f, 0.f, 0.f, 0.f};
#pragma unroll
        for (int kk = 0; kk < 16; ++kk) {
            v2f bf;
            bf.x = Wl[(kk * 4 + kp) * 65 + col];
            bf.y = Wl[(kk * 4 + kp + 1) * 65 + col];
            acc = __builtin_amdgcn_wmma_f32_16x16x4_f32(false, afrag[kk],
                                                        false, bf,
                                                        (short)0, acc, false, false);
        }
        float av = Al[nt * 16 + (lane & 15)];
#pragma unroll
        for (int r = 0; r < 8; ++r) {
            int rowOut = tile * 16 + r + ((lane >> 4) << 3);
            float t = acc[r] * av;
            t += __shfl_xor(t, 1, 32);
            t += __shfl_xor(t, 2, 32);
            t += __shfl_xor(t, 4, 32);
            t += __shfl_xor(t, 8, 32);           // per-16-lane-half row sum
            if (rowOut < n) {
                if (Xout) Xout[(size_t)rowOut * 64 + col] = acc[r];
                if ((lane & 15) == 0) Aout[(size_t)rowOut * 4 + nt] = t;
            }
        }
    }
}

// ---------------- edge-softmax passes ----------------

__device__ __forceinline__ float atomicMaxF(float* addr, float val) {
    if (val >= 0.0f)
        return __int_as_float(atomicMax((int*)addr, __float_as_int(val)));
    else
        return __uint_as_float(atomicMin((unsigned int*)addr, __float_as_uint(val)));
}

__device__ __forceinline__ float lrelu(float v) { return (v >= 0.f) ? v : 0.2f * v; }

__global__ void k_edge_max(const int* __restrict__ src, const int* __restrict__ dst,
                           const float* __restrict__ as_, const float* __restrict__ ad_,
                           float* __restrict__ smax, int E_) {
    int e = blockIdx.x * blockDim.x + threadIdx.x;
    if (e >= E_) return;
    int s = src[e], d = dst[e];
    const float4 a = *(const float4*)(as_ + (size_t)s * 4);
    const float4 b = *(const float4*)(ad_ + (size_t)d * 4);
    atomicMaxF(&smax[(size_t)d * 4 + 0], lrelu(a.x + b.x));
    atomicMaxF(&smax[(size_t)d * 4 + 1], lrelu(a.y + b.y));
    atomicMaxF(&smax[(size_t)d * 4 + 2], lrelu(a.z + b.z));
    atomicMaxF(&smax[(size_t)d * 4 + 3], lrelu(a.w + b.w));
}

__global__ void k_edge_expsum(const int* __restrict__ src, const int* __restrict__ dst,
                              const float* __restrict__ as_, const float* __restrict__ ad_,
                              const float* __restrict__ smax, float* __restrict__ ssum,
                              float* __restrict__ exbuf, int E_) {
    int e = blockIdx.x * blockDim.x + threadIdx.x;
    if (e >= E_) return;
    int s = src[e], d = dst[e];
    const float4 a = *(const float4*)(as_ + (size_t)s * 4);
    const float4 b = *(const float4*)(ad_ + (size_t)d * 4);
    const float4 m = *(const float4*)(smax + (size_t)d * 4);
    float4 ex;
    ex.x = expf(lrelu(a.x + b.x) - m.x);
    ex.y = expf(lrelu(a.y + b.y) - m.y);
    ex.z = expf(lrelu(a.z + b.z) - m.z);
    ex.w = expf(lrelu(a.w + b.w) - m.w);
    *(float4*)(exbuf + (size_t)e * 4) = ex;
    atomicAdd(&ssum[(size_t)d * 4 + 0], ex.x);
    atomicAdd(&ssum[(size_t)d * 4 + 1], ex.y);
    atomicAdd(&ssum[(size_t)d * 4 + 2], ex.z);
    atomicAdd(&ssum[(size_t)d * 4 + 3], ex.w);
}

// 32 threads per edge; each thread handles elements (lane, lane+32).
__global__ void k_edge_aggr(const int* __restrict__ src, const int* __restrict__ dst,
                            const float* __restrict__ xs, const float* __restrict__ exbuf,
                            const float* __restrict__ ssum, float* __restrict__ acc, int E_) {
    long long tid = (long long)blockIdx.x * blockDim.x + threadIdx.x;
    int e = (int)(tid >> 5);
    if (e >= E_) return;
    int lane = (int)(tid & 31);
    int s = src[e], d = dst[e];
    int h0 = lane >> 4;                // head for element `lane`      (0/1)
    int h1 = h0 + 2;                   // head for element `lane + 32` (2/3)
    float al0 = exbuf[(size_t)e * 4 + h0] / ssum[(size_t)d * 4 + h0];
    float al1 = exbuf[(size_t)e * 4 + h1] / ssum[(size_t)d * 4 + h1];
    atomicAdd(&acc[(size_t)d * 64 + lane],      al0 * xs[(size_t)s * 64 + lane]);
    atomicAdd(&acc[(size_t)d * 64 + lane + 32], al1 * xs[(size_t)s * 64 + lane + 32]);
}

// h2 = elu(acc + bias1 (+ bias2)) in place.
__global__ void k_finalize(float* __restrict__ acc, const float* __restrict__ b1,
                           const float* __restrict__ b2, long long total) {
    long long t = (long long)blockIdx.x * blockDim.x + threadIdx.x;
    if (t >= total) return;
    int col = (int)(t & 63);
    float v = acc[t] + b1[col] + (b2 ? b2[col] : 0.0f);
    acc[t] = (v > 0.f) ? v : expm1f(v);
}

// Head MLP: one wave per node; lane l = hidden unit l (exactly 32 units).
__global__ void k_head(const float* __restrict__ h2t, const float* __restrict__ W1,
                       const float* __restrict__ b1, const float* __restrict__ W2,
                       const float* __restrict__ b2, float* __restrict__ out, int n) {
    int node = blockIdx.x * (blockDim.x >> 5) + (threadIdx.x >> 5);
    int lane = threadIdx.x & 31;
    if (node >= n) return;
    const float* hrow = h2t + (size_t)node * 64;
    float acc = b1[lane];
#pragma unroll 8
    for (int k = 0; k < 64; ++k) acc += hrow[k] * W1[k * 32 + lane];
    acc = fmaxf(acc, 0.0f);
    float p = acc * W2[lane];
    for (int m = 16; m; m >>= 1) p += __shfl_xor(p, m, 32);
    if (lane == 0) out[node] = p + b2[0];
}

// ---------------- host launch ----------------

static inline int cdiv(long long a, long long b) { return (int)((a + b - 1) / b); }

extern "C" void kernel_launch(void* const* d_in, const int* in_sizes, int n_in,
                              void* d_out, int out_size, void* d_ws, size_t ws_size,
                              hipStream_t stream) {
    const int n_w = in_sizes[0];          // x_wallet flat count (N_W x 1)
    const int n_t = in_sizes[1];          // x_token  flat count (N_T x 1)
    const int E_  = in_sizes[2];

    const int* e_wt_src = (const int*)d_in[2];
    const int* e_wt_dst = (const int*)d_in[3];
    const int* e_tw_src = (const int*)d_in[4];
    const int* e_tw_dst = (const int*)d_in[5];
    const int* e_ww_src = (const int*)d_in[6];
    const int* e_ww_dst = (const int*)d_in[7];

    // params flattened via jax tree_leaves (alphabetical keys, nested):
    const float* emb_t    = (const float*)d_in[8];
    const float* emb_w    = (const float*)d_in[9];
    const float* tw_Wdst  = (const float*)d_in[10];
    const float* tw_Wsrc  = (const float*)d_in[11];
    const float* tw_adst  = (const float*)d_in[12];
    const float* tw_asrc  = (const float*)d_in[13];
    const float* tw_bias  = (const float*)d_in[14];
    const float* wt_Wdst  = (const float*)d_in[15];
    const float* wt_Wsrc  = (const float*)d_in[16];
    const float* wt_adst  = (const float*)d_in[17];
    const float* wt_asrc  = (const float*)d_in[18];
    const float* wt_bias  = (const float*)d_in[19];
    const float* ww_Wdst  = (const float*)d_in[20];
    const float* ww_Wsrc  = (const float*)d_in[21];
    const float* ww_adst  = (const float*)d_in[22];
    const float* ww_asrc  = (const float*)d_in[23];
    const float* ww_bias  = (const float*)d_in[24];
    const float* hW1      = (const float*)d_in[25];
    const float* hW2      = (const float*)d_in[26];
    const float* hb1      = (const float*)d_in[27];
    const float* hb2      = (const float*)d_in[28];
    const float* lnT_b    = (const float*)d_in[29];
    const float* lnT_g    = (const float*)d_in[30];
    const float* lnW_b    = (const float*)d_in[31];
    const float* lnW_g    = (const float*)d_in[32];
    const float* sTW_Wl   = (const float*)d_in[33];
    const float* sTW_Wr   = (const float*)d_in[34];
    const float* sTW_bl   = (const float*)d_in[35];
    const float* sWW_Wl   = (const float*)d_in[36];
    const float* sWW_Wr   = (const float*)d_in[37];
    const float* sWW_bl   = (const float*)d_in[38];
    const float* sWT_Wl   = (const float*)d_in[39];
    const float* sWT_Wr   = (const float*)d_in[40];
    const float* sWT_bl   = (const float*)d_in[41];

    // Output layout: logits [n_t], h2_t [n_t*64], h2_w [n_w*64]
    float* out    = (float*)d_out;
    float* logits = out;
    float* acc_t  = out + n_t;                          // h2_t accumulator (in place)
    float* acc_w  = out + n_t + (size_t)n_t * 64;       // h2_w accumulator (in place)

    // Workspace carve-up (16B aligned regions)
    float* ws = (float*)d_ws;
    size_t off = 0;
    auto alloc = [&](size_t cnt) { float* p = ws + off; off += (cnt + 3) & ~(size_t)3; return p; };
    float* vecs   = alloc(320);
    float* deg_wt = alloc((size_t)n_t);
    float* deg_tw = alloc((size_t)n_w);
    float* deg_ww = alloc((size_t)n_w);
    float* h_t    = alloc((size_t)n_t * 64);
    float* h_w    = alloc((size_t)n_w * 64);
    float* xs_wt  = alloc((size_t)n_w * 64);   // src projection (wallets) for wt
    float* xs_tw  = alloc((size_t)n_t * 64);   // src projection (tokens)  for tw
    float* xs_ww  = alloc((size_t)n_w * 64);   // src projection (wallets) for ww
    float* as_wt  = alloc((size_t)n_w * 4);
    float* ad_wt  = alloc((size_t)n_t * 4);
    float* as_tw  = alloc((size_t)n_t * 4);
    float* ad_tw  = alloc((size_t)n_w * 4);
    float* as_ww  = alloc((size_t)n_w * 4);
    float* ad_ww  = alloc((size_t)n_w * 4);
    int n_max = (n_t > n_w) ? n_t : n_w;
    float* smax   = alloc((size_t)n_max * 4);
    float* ssum   = alloc((size_t)n_max * 4);
    float* exbuf  = alloc((size_t)E_ * 4);

    const int TB = 256;

    // 0. zero accumulators & degree buffers
    k_fill<<<cdiv((long long)n_t * 64, TB), TB, 0, stream>>>(acc_t, 0.0f, (long long)n_t * 64);
    k_fill<<<cdiv((long long)n_w * 64, TB), TB, 0, stream>>>(acc_w, 0.0f, (long long)n_w * 64);
    k_fill<<<cdiv(n_t, TB), TB, 0, stream>>>(deg_wt, 0.0f, n_t);
    k_fill<<<cdiv(n_w, TB), TB, 0, stream>>>(deg_tw, 0.0f, n_w);
    k_fill<<<cdiv(n_w, TB), TB, 0, stream>>>(deg_ww, 0.0f, n_w);

    // 1. tiny matvecs + degrees
    k_small_vecs<<<1, 64, 0, stream>>>(emb_w, emb_t,
                                       sWT_Wl, sWT_Wr, sWT_bl,
                                       sTW_Wl, sTW_Wr, sTW_bl,
                                       sWW_Wl, sWW_Wr, sWW_bl, vecs);
    k_degree<<<cdiv(E_, TB), TB, 0, stream>>>(e_wt_dst, deg_wt, E_);
    k_degree<<<cdiv(E_, TB), TB, 0, stream>>>(e_tw_dst, deg_tw, E_);
    k_degree<<<cdiv(E_, TB), TB, 0, stream>>>(e_ww_dst, deg_ww, E_);

    // 2. layer 1: LN + ELU
    k_layer1<<<cdiv(n_t, 8), TB, 0, stream>>>(deg_wt, nullptr, vecs, nullptr, vecs + 64,
                                              lnT_g, lnT_b, h_t, n_t);
    k_layer1<<<cdiv(n_w, 8), TB, 0, stream>>>(deg_tw, deg_ww, vecs + 128, vecs + 192, vecs + 256,
                                              lnW_g, lnW_b, h_w, n_w);

    // 3. WMMA projections + fused attention scores
    int gbT = cdiv(cdiv(n_t, 16), 4);
    int gbW = cdiv(cdiv(n_w, 16), 4);
    k_project<<<gbW, 128, 0, stream>>>(h_w, wt_Wsrc, wt_asrc, xs_wt,  as_wt, n_w);
    k_project<<<gbT, 128, 0, stream>>>(h_t, wt_Wdst, wt_adst, nullptr, ad_wt, n_t);
    k_project<<<gbT, 128, 0, stream>>>(h_t, tw_Wsrc, tw_asrc, xs_tw,  as_tw, n_t);
    k_project<<<gbW, 128, 0, stream>>>(h_w, tw_Wdst, tw_adst, nullptr, ad_tw, n_w);
    k_project<<<gbW, 128, 0, stream>>>(h_w, ww_Wsrc, ww_asrc, xs_ww,  as_ww, n_w);
    k_project<<<gbW, 128, 0, stream>>>(h_w, ww_Wdst, ww_adst, nullptr, ad_ww, n_w);

    // 4. edge softmax + aggregation, one edge type at a time (stream-ordered)
    // --- wallet -> token ---
    k_fill<<<cdiv((long long)n_t * 4, TB), TB, 0, stream>>>(smax, -INFINITY, (long long)n_t * 4);
    k_fill<<<cdiv((long long)n_t * 4, TB), TB, 0, stream>>>(ssum, 0.0f, (long long)n_t * 4);
    k_edge_max<<<cdiv(E_, TB), TB, 0, stream>>>(e_wt_src, e_wt_dst, as_wt, ad_wt, smax, E_);
    k_edge_expsum<<<cdiv(E_, TB), TB, 0, stream>>>(e_wt_src, e_wt_dst, as_wt, ad_wt, smax, ssum, exbuf, E_);
    k_edge_aggr<<<cdiv((long long)E_ * 32, TB), TB, 0, stream>>>(e_wt_src, e_wt_dst, xs_wt, exbuf, ssum, acc_t, E_);
    // --- token -> wallet ---
    k_fill<<<cdiv((long long)n_w * 4, TB), TB, 0, stream>>>(smax, -INFINITY, (long long)n_w * 4);
    k_fill<<<cdiv((long long)n_w * 4, TB), TB, 0, stream>>>(ssum, 0.0f, (long long)n_w * 4);
    k_edge_max<<<cdiv(E_, TB), TB, 0, stream>>>(e_tw_src, e_tw_dst, as_tw, ad_tw, smax, E_);
    k_edge_expsum<<<cdiv(E_, TB), TB, 0, stream>>>(e_tw_src, e_tw_dst, as_tw, ad_tw, smax, ssum, exbuf, E_);
    k_edge_aggr<<<cdiv((long long)E_ * 32, TB), TB, 0, stream>>>(e_tw_src, e_tw_dst, xs_tw, exbuf, ssum, acc_w, E_);
    // --- wallet -> wallet ---
    k_fill<<<cdiv((long long)n_w * 4, TB), TB, 0, stream>>>(smax, -INFINITY, (long long)n_w * 4);
    k_fill<<<cdiv((long long)n_w * 4, TB), TB, 0, stream>>>(ssum, 0.0f, (long long)n_w * 4);
    k_edge_max<<<cdiv(E_, TB), TB, 0, stream>>>(e_ww_src, e_ww_dst, as_ww, ad_ww, smax, E_);
    k_edge_expsum<<<cdiv(E_, TB), TB, 0, stream>>>(e_ww_src, e_ww_dst, as_ww, ad_ww, smax, ssum, exbuf, E_);
    k_edge_aggr<<<cdiv((long long)E_ * 32, TB), TB, 0, stream>>>(e_ww_src, e_ww_dst, xs_ww, exbuf, ssum, acc_w, E_);

    // 5. bias + ELU (in place in d_out)
    k_finalize<<<cdiv((long long)n_t * 64, TB), TB, 0, stream>>>(acc_t, wt_bias, nullptr, (long long)n_t * 64);
    k_finalize<<<cdiv((long long)n_w * 64, TB), TB, 0, stream>>>(acc_w, tw_bias, ww_bias, (long long)n_w * 64);

    // 6. head MLP -> logits
    k_head<<<cdiv(n_t, 8), TB, 0, stream>>>(acc_t, hW1, hb1, hW2, hb2, logits, n_t);

    (void)n_in; (void)out_size; (void)ws_size;
}